// WisePooling_5239860101875
// MI455X (gfx1250) — compile-verified
//
#include <hip/hip_runtime.h>
#include <hip/hip_bf16.h>

// Segment-mean pooling (ragged over time) for x[B=16, T=8192, D=256] fp32,
// graph[B, S=64, 2] int32 -> out[B, S, D] fp32.
//
// MI455X strategy:
//  - Bandwidth-bound: 134 MB in / 1 MB out -> ~5.8 us floor at 23.3 TB/s.
//  - Reduction engine is V_WMMA_F32_16X16X4_F32 with A = ones(16x4):
//      D[m,n] = sum_k B[k,n] + C[m,n]   (all rows identical; row M=0 used).
//    Each B slot pre-reduces 2 timesteps with one v_add_f32, so one WMMA
//    consumes 8 timesteps x 16 features; f32 matrix-pipe demand stays far
//    below its ceiling and HBM remains the limiter.
//  - One wave32 per (b, s, 16-feature tile): 16*64*16 = 16384 waves.
//  - readfirstlane on [start,end] -> scalar loop control (s_cbranch_scc).
//  - Manual x2 unroll with two accumulators: 8 clause-grouped loads in
//    flight per iteration; accumulators merged with one v_add_f32 at the end.
//  - Ragged tail (<8 rows, wave-uniform branch): address-clamped loads with
//    value masking, keeping EXEC all-1s for the WMMA.

typedef __attribute__((ext_vector_type(2))) float v2f;
typedef __attribute__((ext_vector_type(8))) float v8f;

#define BB 16
#define TT 8192
#define DD 256
#define SS 64
#define EPS_ADD 0.006f
#define WAVES_PER_BLOCK 8

__global__ __launch_bounds__(256)
void WisePooling_seg_mean_wmma(const float* __restrict__ x,
                               const int*   __restrict__ graph,
                               float*       __restrict__ out) {
    const int lane = threadIdx.x & 31;
    const int wave = threadIdx.x >> 5;

    const int tile  = blockIdx.x * WAVES_PER_BLOCK + wave;  // 0 .. 16383
    const int dtile = tile & 15;                            // feature tile of 16
    const int bs    = tile >> 4;                            // 0 .. B*S-1
    const int b     = bs >> 6;                              // / SS

    // Wave-uniform segment bounds -> SGPRs, so the loop is scalar-controlled.
    const int t_start = __builtin_amdgcn_readfirstlane(graph[bs * 2 + 0]);
    const int t_end   = __builtin_amdgcn_readfirstlane(graph[bs * 2 + 1]);
    const int len     = t_end - t_start + 1;

    const int d0    = dtile * 16;
    const int khalf = (lane >> 4) << 1;     // 0 (lanes 0-15) or 2 (lanes 16-31)

    // Column pointer for this lane's feature; row t lives at colf[t * DD].
    const float* __restrict__ colf =
        x + (size_t)b * TT * DD + d0 + (lane & 15);

    // Main-loop pointer with t_start and khalf folded in; all loads below use
    // 24-bit immediate offsets (0..13*DD floats = 0..13312 bytes).
    const float* __restrict__ p = colf + (size_t)(t_start + khalf) * DD;

    v2f a_ones; a_ones.x = 1.0f; a_ones.y = 1.0f;
    v8f acc0 = {};
    v8f acc1 = {};

    const int nchunks = len >> 3;           // full 8-timestep chunks
    int i = 0;
    for (; i + 1 < nchunks; i += 2) {       // 16 timesteps / iteration
        v2f b0, b1;
        b0.x = p[0]       + p[4 * DD];      // rows +0, +4   (K slot pair 0)
        b0.y = p[DD]      + p[5 * DD];      // rows +1, +5
        b1.x = p[8 * DD]  + p[12 * DD];     // rows +8, +12
        b1.y = p[9 * DD]  + p[13 * DD];     // rows +9, +13
        acc0 = __builtin_amdgcn_wmma_f32_16x16x4_f32(
            false, a_ones, false, b0, (short)0, acc0, false, false);
        acc1 = __builtin_amdgcn_wmma_f32_16x16x4_f32(
            false, a_ones, false, b1, (short)0, acc1, false, false);
        p += 16 * DD;
    }
    if (i < nchunks) {                      // odd leftover full chunk
        v2f b0;
        b0.x = p[0]  + p[4 * DD];
        b0.y = p[DD] + p[5 * DD];
        acc0 = __builtin_amdgcn_wmma_f32_16x16x4_f32(
            false, a_ones, false, b0, (short)0, acc0, false, false);
    }

    // Ragged tail: < 8 rows. Wave-uniform branch (scalar t0/t_end), EXEC
    // stays all-1s for the WMMA. Addresses clamped in-range; out-of-range
    // slots contribute 0 via value selects.
    const int t0 = t_start + (nchunks << 3);
    if (t0 <= t_end) {
        const int ta = t0 + khalf;
        const int tb = ta + 1;
        v2f bm;
        {
            int u0 = ta,     c0 = (u0 <= t_end) ? u0 : t_end;
            int u1 = ta + 4, c1 = (u1 <= t_end) ? u1 : t_end;
            float r0 = colf[(size_t)c0 * DD];
            float r1 = colf[(size_t)c1 * DD];
            bm.x = ((u0 <= t_end) ? r0 : 0.0f) + ((u1 <= t_end) ? r1 : 0.0f);
        }
        {
            int u0 = tb,     c0 = (u0 <= t_end) ? u0 : t_end;
            int u1 = tb + 4, c1 = (u1 <= t_end) ? u1 : t_end;
            float r0 = colf[(size_t)c0 * DD];
            float r1 = colf[(size_t)c1 * DD];
            bm.y = ((u0 <= t_end) ? r0 : 0.0f) + ((u1 <= t_end) ? r1 : 0.0f);
        }
        acc0 = __builtin_amdgcn_wmma_f32_16x16x4_f32(
            false, a_ones, false, bm, (short)0, acc0, false, false);
    }

    const float inv_len = 1.0f / (float)len;
    // Row M=0 lives in component 0, lanes 0-15 (N = lane). Merge the two
    // accumulators on just that component.
    const float sum = acc0[0] + acc1[0];
    if (lane < 16) {
        out[(size_t)bs * DD + d0 + lane] = sum * inv_len + EPS_ADD;
    }
}

extern "C" void kernel_launch(void* const* d_in, const int* in_sizes, int n_in,
                              void* d_out, int out_size, void* d_ws, size_t ws_size,
                              hipStream_t stream) {
    (void)in_sizes; (void)n_in; (void)out_size; (void)d_ws; (void)ws_size;
    const float* x     = (const float*)d_in[0];
    const int*   graph = (const int*)d_in[1];
    float*       out   = (float*)d_out;

    const int total_tiles = BB * SS * 16;               // 16384 waves
    const int blocks = total_tiles / WAVES_PER_BLOCK;   // 2048 blocks
    WisePooling_seg_mean_wmma<<<dim3(blocks), dim3(256), 0, stream>>>(x, graph, out);
}